// GCN_39152921870513
// MI455X (gfx1250) — compile-verified
//
#include <hip/hip_runtime.h>

#define N_NODES 1024
#define TOPK    32
#define C1_     92
#define DQ_     256
#define DH_     512
#define DO_     256
#define DMLP_   1024
#define DIN_    352
#define NB_     64
#define CB_     4     // batch chunk

typedef float v2f __attribute__((ext_vector_type(2)));
typedef float v8f __attribute__((ext_vector_type(8)));

// ---------------------------------------------------------------------------
// Generic fp32 WMMA GEMM: C[z] = A[z] (MxK, row-major) * B[z] (KxN, row-major)
// Block tile 128x128, K-stage 32, 8 waves (wave32); each wave computes 32x64
// (2 A-frags reuse 4 B-frags -> 1.5 LDS dwords per WMMA) with
// V_WMMA_F32_16X16X4_F32. Software-pipelined: next K-tile is fetched into
// registers (b128 loads) while the current tile's WMMAs execute from LDS.
// ---------------------------------------------------------------------------
template<bool BIAS, bool RELU>
__global__ __launch_bounds__(256) void gemm_f32_wmma(
    const float* __restrict__ A, const float* __restrict__ B,
    const float* __restrict__ bias, float* __restrict__ C,
    int Mdim, int Ndim, int Kdim,
    long sA, long sB, long sC)
{
  constexpr int BM = 128, BN = 128, KC = 32;
  __shared__ float As[KC][BM + 1];   // stride 129 (==1 mod 64): conflict-free
  __shared__ float Bs[KC][BN + 16];  // stride 144: disjoint half-wave banks

  A += (long)blockIdx.z * sA;
  B += (long)blockIdx.z * sB;
  C += (long)blockIdx.z * sC;

  const int m0   = blockIdx.y * BM;
  const int n0   = blockIdx.x * BN;
  const int tid  = threadIdx.x;
  const int lane = tid & 31;
  const int wave = tid >> 5;
  const int wm   = wave >> 1;          // 0..3 -> 32-row sub-tile
  const int wn   = wave & 1;           // 0..1 -> 64-col sub-tile
  const int lm   = lane & 15;
  const int lk   = (lane >> 4) * 2;    // 0 or 2 (A/B frag K halves)

  // staging maps (all b128, 16B-aligned: K,N dims are multiples of 4/128)
  const int amA = tid >> 3;            // A row (j adds 32)
  const int akq = (tid & 7) * 4;       // A k-quad
  const int bk  = tid >> 5;            // B k row (j adds 8)
  const int bnq = (tid & 31) * 4;      // B n-quad

  float4 ra[4], rb[4];

  auto gload = [&](int k0) {
    #pragma unroll
    for (int j = 0; j < 4; ++j)
      ra[j] = *(const float4*)&A[(size_t)(m0 + amA + 32 * j) * Kdim + k0 + akq];
    #pragma unroll
    for (int j = 0; j < 4; ++j)
      rb[j] = *(const float4*)&B[(size_t)(k0 + bk + 8 * j) * Ndim + n0 + bnq];
  };
  auto sstore = [&]() {
    #pragma unroll
    for (int j = 0; j < 4; ++j) {
      As[akq + 0][amA + 32 * j] = ra[j].x;
      As[akq + 1][amA + 32 * j] = ra[j].y;
      As[akq + 2][amA + 32 * j] = ra[j].z;
      As[akq + 3][amA + 32 * j] = ra[j].w;
    }
    #pragma unroll
    for (int j = 0; j < 4; ++j)
      *(float4*)&Bs[bk + 8 * j][bnq] = rb[j];
  };

  v8f acc[2][4] = {};

  gload(0);
  for (int k0 = 0; k0 < Kdim; k0 += KC) {
    __syncthreads();                       // prior MMA reads complete
    sstore();
    __syncthreads();                       // tile visible
    if (k0 + KC < Kdim) gload(k0 + KC);    // prefetch overlaps MMA (uniform)

    #pragma unroll
    for (int kk = 0; kk < KC; kk += 4) {
      v2f a0, a1;
      a0.x = As[kk + lk + 0][wm * 32 + lm];
      a0.y = As[kk + lk + 1][wm * 32 + lm];
      a1.x = As[kk + lk + 0][wm * 32 + 16 + lm];
      a1.y = As[kk + lk + 1][wm * 32 + 16 + lm];
      #pragma unroll
      for (int t = 0; t < 4; ++t) {
        v2f b;
        b.x = Bs[kk + lk + 0][wn * 64 + t * 16 + lm];
        b.y = Bs[kk + lk + 1][wn * 64 + t * 16 + lm];
        acc[0][t] = __builtin_amdgcn_wmma_f32_16x16x4_f32(
            false, a0, false, b, (short)0, acc[0][t], false, false);
        acc[1][t] = __builtin_amdgcn_wmma_f32_16x16x4_f32(
            false, a1, false, b, (short)0, acc[1][t], false, false);
      }
    }
  }

  #pragma unroll
  for (int h = 0; h < 2; ++h) {
    const int rbase = m0 + wm * 32 + h * 16 + (lane >> 4) * 8;
    #pragma unroll
    for (int t = 0; t < 4; ++t) {
      const int col = n0 + wn * 64 + t * 16 + lm;
      const float bv = BIAS ? bias[col] : 0.0f;
      #pragma unroll
      for (int r = 0; r < 8; ++r) {
        float v = acc[h][t][r] + bv;
        if (RELU) v = fmaxf(v, 0.0f);
        C[(size_t)(rbase + r) * Ndim + col] = v;
      }
    }
  }
}

// ---------------------------------------------------------------------------
// Build X = concat(query_emb, probs, bbox) for a batch chunk
// ---------------------------------------------------------------------------
__global__ void pack_x(const float* __restrict__ probs, const float* __restrict__ bbox,
                       const float* __restrict__ qemb, float* __restrict__ X, int bbase)
{
  long e = (long)blockIdx.x * blockDim.x + threadIdx.x;
  if (e >= (long)CB_ * N_NODES * DIN_) return;
  int col = (int)(e % DIN_);
  int i   = (int)((e / DIN_) % N_NODES);
  int b   = bbase + (int)(e / ((long)DIN_ * N_NODES));
  float v;
  if (col < DQ_)            v = qemb[(size_t)i * DQ_ + col];
  else if (col < DQ_ + C1_) v = probs[((size_t)b * N_NODES + i) * C1_ + (col - DQ_)];
  else                      v = bbox[((size_t)b * N_NODES + i) * 4 + (col - DQ_ - C1_)];
  X[e] = v;
}

// ---------------------------------------------------------------------------
// Top-32 of 1024 per row: one wave per row, wave-wide argmax x32 (desc order)
// ---------------------------------------------------------------------------
__global__ __launch_bounds__(256) void topk_rows(const float* __restrict__ adj,
    float* __restrict__ vals, int* __restrict__ idxs, long row_base)
{
  const int lane = threadIdx.x & 31;
  const int wv   = threadIdx.x >> 5;
  const long row = (long)blockIdx.x * 8 + wv;
  const float* r = adj + row * N_NODES;

  float v[32];
  #pragma unroll
  for (int t = 0; t < 32; ++t) v[t] = r[t * 32 + lane];

  float lmax = v[0]; int larg = 0;
  #pragma unroll
  for (int t = 1; t < 32; ++t) if (v[t] > lmax) { lmax = v[t]; larg = t; }

  float* vo = vals + (row_base + row) * TOPK;
  int*   io = idxs + (row_base + row) * TOPK;

  for (int k = 0; k < TOPK; ++k) {
    float m = lmax; int gi = larg * 32 + lane;
    #pragma unroll
    for (int off = 16; off > 0; off >>= 1) {
      float om = __shfl_xor(m, off, 32);
      int   oi = __shfl_xor(gi, off, 32);
      if (om > m || (om == m && oi < gi)) { m = om; gi = oi; }
    }
    if (lane == 0) { vo[k] = m; io[k] = gi; }
    const int wl = gi & 31, wt = gi >> 5;
    if (lane == wl) {
      #pragma unroll
      for (int t = 0; t < 32; ++t) if (t == wt) v[t] = -__builtin_inff();
    }
    lmax = v[0]; larg = 0;
    #pragma unroll
    for (int t = 1; t < 32; ++t) if (v[t] > lmax) { lmax = v[t]; larg = t; }
  }
}

// --------------------------- graph normalization ---------------------------
__global__ void deg_init(float* deg, long n) {
  long e = (long)blockIdx.x * blockDim.x + threadIdx.x;
  if (e < n) deg[e] = 1.0f;                  // self-loop weight
}
__global__ void deg_scatter(const float* __restrict__ vals, const int* __restrict__ idxs,
                            float* deg, long row_base) {
  long e = (long)blockIdx.x * blockDim.x + threadIdx.x;
  if (e >= (long)CB_ * N_NODES * TOPK) return;
  long row = e / TOPK;
  int  bl  = (int)(row / N_NODES);
  long ge  = (row_base + row) * TOPK + (e % TOPK);
  atomicAdd(&deg[(long)bl * N_NODES + idxs[ge]], vals[ge]);
}
__global__ void make_dinv(const float* deg, float* dinv, long n) {
  long e = (long)blockIdx.x * blockDim.x + threadIdx.x;
  if (e >= n) return;
  float d = deg[e];
  dinv[e] = (d == 0.0f) ? 0.0f : 1.0f / sqrtf(d);   // NaN for d<0, like ref
}

__global__ void zero_f32(float* p, long n) {
  long e = (long)blockIdx.x * blockDim.x + threadIdx.x;
  if (e < n) p[e] = 0.0f;
}
__global__ void m_diag(float* Mm, const float* dinv, long n) {
  long e = (long)blockIdx.x * blockDim.x + threadIdx.x;
  if (e >= n) return;
  int bl = (int)(e / N_NODES), i = (int)(e % N_NODES);
  float d = dinv[e];
  Mm[(size_t)bl * N_NODES * N_NODES + (size_t)i * N_NODES + i] = d * d;
}
__global__ void m_scatter(const float* __restrict__ vals, const int* __restrict__ idxs,
                          const float* __restrict__ dinv, float* Mm, long row_base) {
  long e = (long)blockIdx.x * blockDim.x + threadIdx.x;
  if (e >= (long)CB_ * N_NODES * TOPK) return;
  long row = e / TOPK;
  int  bl  = (int)(row / N_NODES);
  int  i   = (int)(row % N_NODES);
  long ge  = (row_base + row) * TOPK + (e % TOPK);
  int  j   = idxs[ge];
  float w  = vals[ge];
  atomicAdd(&Mm[(size_t)bl * N_NODES * N_NODES + (size_t)j * N_NODES + i],
            dinv[(long)bl * N_NODES + i] * w * dinv[(long)bl * N_NODES + j]);
}

// ------------------------------- batch norm --------------------------------
__global__ __launch_bounds__(256) void bn_stats(const float* __restrict__ h,
                                                float* mu, float* rinv) {
  int bl = blockIdx.x, t = threadIdx.x;
  const float* hb = h + (size_t)bl * N_NODES * DH_;
  float s0 = 0, q0 = 0, s1 = 0, q1 = 0;
  for (int i = 0; i < N_NODES; ++i) {
    float a = hb[(size_t)i * DH_ + t];
    float b = hb[(size_t)i * DH_ + t + 256];
    s0 += a; q0 += a * a; s1 += b; q1 += b * b;
  }
  float m0 = s0 / N_NODES, m1 = s1 / N_NODES;
  mu[bl * DH_ + t]        = m0;
  mu[bl * DH_ + t + 256]  = m1;
  rinv[bl * DH_ + t]       = 1.0f / sqrtf(q0 / N_NODES - m0 * m0 + 1e-5f);
  rinv[bl * DH_ + t + 256] = 1.0f / sqrtf(q1 / N_NODES - m1 * m1 + 1e-5f);
}
__global__ void bn_apply(float* h, const float* mu, const float* rinv,
                         const float* gamma, const float* beta, long n) {
  long e = (long)blockIdx.x * blockDim.x + threadIdx.x;
  if (e >= n) return;
  int c  = (int)(e % DH_);
  int bl = (int)(e / ((long)N_NODES * DH_));
  float x = h[e];
  x = (x - mu[bl * DH_ + c]) * rinv[bl * DH_ + c] * gamma[c] + beta[c];
  h[e] = fmaxf(x, 0.0f);
}

// ---------------------------------------------------------------------------
extern "C" void kernel_launch(void* const* d_in, const int* in_sizes, int n_in,
                              void* d_out, int out_size, void* d_ws, size_t ws_size,
                              hipStream_t stream)
{
  const float* probs = (const float*)d_in[0];
  const float* bbox  = (const float*)d_in[1];
  const float* qemb  = (const float*)d_in[2];
  const float* nemb  = (const float*)d_in[3];
  const float* w1    = (const float*)d_in[4];
  const float* b1    = (const float*)d_in[5];
  const float* w2    = (const float*)d_in[6];
  const float* b2    = (const float*)d_in[7];
  const float* c1w   = (const float*)d_in[8];
  const float* c1b   = (const float*)d_in[9];
  const float* c2w   = (const float*)d_in[10];
  const float* c2b   = (const float*)d_in[11];
  const float* bng   = (const float*)d_in[12];
  const float* bnb   = (const float*)d_in[13];
  float* out = (float*)d_out;

  char* p = (char*)d_ws;
  auto carve = [&](size_t bytes) { char* q = p; p += (bytes + 255) & ~(size_t)255; return q; };
  float* vals = (float*)carve((size_t)NB_ * N_NODES * TOPK * 4);
  int*   idxs = (int*)  carve((size_t)NB_ * N_NODES * TOPK * 4);
  float* xw1  = (float*)carve((size_t)N_NODES * DH_ * 4);
  float* XB   = (float*)carve((size_t)CB_ * N_NODES * DIN_ * 4);
  float* H1   = (float*)carve((size_t)CB_ * N_NODES * DMLP_ * 4);    // reused as M
  float* ADJ  = (float*)carve((size_t)CB_ * N_NODES * N_NODES * 4);  // reused as h,T2
  float* deg  = (float*)carve((size_t)CB_ * N_NODES * 4);
  float* dinv = (float*)carve((size_t)CB_ * N_NODES * 4);
  float* mu   = (float*)carve((size_t)CB_ * DH_ * 4);
  float* rinv = (float*)carve((size_t)CB_ * DH_ * 4);
  float* Mm   = H1;
  float* hbuf = ADJ;
  float* T2   = ADJ + (size_t)CB_ * N_NODES * DH_;

  // xw1 = node_emb @ conv1_w   (1024x256 @ 256x512), shared across batch
  gemm_f32_wmma<false, false><<<dim3(DH_ / 128, N_NODES / 128, 1), 256, 0, stream>>>(
      nemb, c1w, nullptr, xw1, N_NODES, DH_, DQ_, 0, 0, 0);

  for (int base = 0; base < NB_; base += CB_) {
    const long row_base = (long)base * N_NODES;

    { long n = (long)CB_ * N_NODES * DIN_;
      pack_x<<<(n + 255) / 256, 256, 0, stream>>>(probs, bbox, qemb, XB, base); }

    // H1 = relu(X @ w1 + b1)
    gemm_f32_wmma<true, true><<<dim3(DMLP_ / 128, CB_ * N_NODES / 128, 1), 256, 0, stream>>>(
        XB, w1, b1, H1, CB_ * N_NODES, DMLP_, DIN_, 0, 0, 0);
    // ADJ = H1 @ w2 + b2
    gemm_f32_wmma<true, false><<<dim3(N_NODES / 128, CB_ * N_NODES / 128, 1), 256, 0, stream>>>(
        H1, w2, b2, ADJ, CB_ * N_NODES, N_NODES, DMLP_, 0, 0, 0);

    topk_rows<<<CB_ * N_NODES / 8, 256, 0, stream>>>(ADJ, vals, idxs, row_base);

    { long n = (long)CB_ * N_NODES;
      deg_init<<<(n + 255) / 256, 256, 0, stream>>>(deg, n); }
    { long n = (long)CB_ * N_NODES * TOPK;
      deg_scatter<<<(n + 255) / 256, 256, 0, stream>>>(vals, idxs, deg, row_base); }
    { long n = (long)CB_ * N_NODES;
      make_dinv<<<(n + 255) / 256, 256, 0, stream>>>(deg, dinv, n); }

    { long n = (long)CB_ * N_NODES * N_NODES;
      zero_f32<<<(n + 255) / 256, 256, 0, stream>>>(Mm, n); }
    { long n = (long)CB_ * N_NODES;
      m_diag<<<(n + 255) / 256, 256, 0, stream>>>(Mm, dinv, n); }
    { long n = (long)CB_ * N_NODES * TOPK;
      m_scatter<<<(n + 255) / 256, 256, 0, stream>>>(vals, idxs, dinv, Mm, row_base); }

    // h = M @ xw1 + conv1_b   (batched over CB_)
    gemm_f32_wmma<true, false><<<dim3(DH_ / 128, N_NODES / 128, CB_), 256, 0, stream>>>(
        Mm, xw1, c1b, hbuf, N_NODES, DH_, N_NODES,
        (long)N_NODES * N_NODES, 0, (long)N_NODES * DH_);

    bn_stats<<<CB_, 256, 0, stream>>>(hbuf, mu, rinv);
    { long n = (long)CB_ * N_NODES * DH_;
      bn_apply<<<(n + 255) / 256, 256, 0, stream>>>(hbuf, mu, rinv, bng, bnb, n); }

    // T2 = h @ conv2_w   (batched, shared B)
    gemm_f32_wmma<false, false><<<dim3(DO_ / 128, N_NODES / 128, CB_), 256, 0, stream>>>(
        hbuf, c2w, nullptr, T2, N_NODES, DO_, DH_,
        (long)N_NODES * DH_, 0, (long)N_NODES * DO_);

    // out = M @ T2 + conv2_b
    gemm_f32_wmma<true, false><<<dim3(DO_ / 128, N_NODES / 128, CB_), 256, 0, stream>>>(
        Mm, T2, c2b, out + row_base * DO_, N_NODES, DO_, N_NODES,
        (long)N_NODES * N_NODES, (long)N_NODES * DO_, (long)N_NODES * DO_);
  }
}